// Bounce1D_83769042141456
// MI455X (gfx1250) — compile-verified
//
#include <hip/hip_runtime.h>
#include <math.h>

typedef __attribute__((ext_vector_type(2))) float v2f;
typedef __attribute__((ext_vector_type(8))) float v8f;

#define Z_N 800
#define R_N 8
#define A_N 16
#define P_N 48
#define W_N 32
#define Q_N 32
#define CAP 512
#define BIGF 1.0e9f
#define PI_F 3.14159265358979323846f

// ---------------------------------------------------------------------------
// Setup: 32-point Gauss-Legendre nodes/weights (Newton on P_32, f64), with the
// sin(pi*x/2) change of variables applied, written to workspace.
// ws layout: [0..31] = xs (transformed nodes), [32..63] = ws (transformed wts)
// ---------------------------------------------------------------------------
__global__ __launch_bounds__(32) void gl_setup_kernel(float* __restrict__ wsbuf) {
    const int i = threadIdx.x;           // 0..31
    const int n = 32;
    double x = cos(M_PI * ((double)i + 0.75) / ((double)n + 0.5));
    double pp = 1.0;
    for (int it = 0; it < 64; ++it) {    // fixed iteration count: deterministic
        double p0 = 1.0, p1 = x;
        for (int k = 2; k <= n; ++k) {
            double pk = ((2.0 * k - 1.0) * x * p1 - (k - 1.0) * p0) / (double)k;
            p0 = p1; p1 = pk;
        }
        pp = (double)n * (x * p0 - p1) / (x * x - 1.0);
        x -= p1 / pp;
    }
    double w = 2.0 / ((1.0 - x * x) * pp * pp);
    wsbuf[i]      = (float)sin(M_PI * x * 0.5);
    wsbuf[32 + i] = (float)(w * (M_PI * 0.5) * cos(M_PI * x * 0.5));
}

// ---------------------------------------------------------------------------
// Main kernel: one workgroup per (r, a, p).
// ---------------------------------------------------------------------------
__global__ __launch_bounds__(256) void bounce_kernel(
    const float* __restrict__ zeta,
    const float* __restrict__ B,
    const float* __restrict__ B_z,
    const float* __restrict__ Bsup,
    const float* __restrict__ Bsup_z,
    const float* __restrict__ glbuf,
    float* __restrict__ out)
{
    __shared__ float sz[Z_N];
    __shared__ float sB[Z_N];
    __shared__ float sBz[Z_N];
    __shared__ float sbv[Z_N];
    __shared__ float sbz[Z_N];
    __shared__ float rz[CAP];
    __shared__ float rd[CAP];
    __shared__ float Gmat[W_N * Q_N];
    __shared__ float z1s[W_N];
    __shared__ float z2s[W_N];
    __shared__ float sxs[Q_N];
    __shared__ float sws[Q_N];
    __shared__ int   nRoots;

    const int tid = threadIdx.x;
    const int blk = blockIdx.x;
    const int p   = blk % P_N;
    const int a   = (blk / P_N) % A_N;
    const int r   = blk / (P_N * A_N);
    const int off = (r * A_N + a) * Z_N;

    // ---- Phase 1: stage line data into LDS, derive b = |Bsup|/B and b' -----
    float lmin =  3.4e38f, lmax = -3.4e38f;
    for (int i = tid; i < Z_N; i += 256) {
        float zv   = zeta[i];
        float Bi   = B[off + i];
        float Bzi  = B_z[off + i];
        float Si   = Bsup[off + i];
        float Szi  = Bsup_z[off + i];
        float sgn  = (Si > 0.0f) ? 1.0f : ((Si < 0.0f) ? -1.0f : 0.0f);
        float bi   = fabsf(Si) / Bi;
        sz[i]  = zv;
        sB[i]  = Bi;
        sBz[i] = Bzi;
        sbv[i] = bi;
        sbz[i] = (Szi * sgn - bi * Bzi) / Bi;
        lmin = fminf(lmin, Bi);
        lmax = fmaxf(lmax, Bi);
    }
    if (tid < Q_N) {
        sxs[tid] = glbuf[tid];
        sws[tid] = glbuf[32 + tid];
    }
    // ---- Phase 2: min/max reduction over B (reuse rz/rd as scratch) --------
    rz[tid] = lmin;
    rd[tid] = lmax;
    __syncthreads();
    for (int s = 128; s > 0; s >>= 1) {
        if (tid < s) {
            rz[tid] = fminf(rz[tid], rz[tid + s]);
            rd[tid] = fmaxf(rd[tid], rd[tid + s]);
        }
        __syncthreads();
    }
    const float minB = rz[0];
    const float maxB = rd[0];
    __syncthreads();

    const float xk  = -1.0f + (2.0f * (float)p + 1.0f) / (float)P_N;
    const float piv = minB + 0.5f * (maxB - minB) * (xk + 1.0f);
    const float invh0 = 1.0f / (sz[1] - sz[0]);

    if (tid == 0) nRoots = 0;
    if (tid < W_N) { z1s[tid] = BIGF; z2s[tid] = BIGF; }
    __syncthreads();

    // ---- Phase 3: cubic root finding per interval (reference formulas) -----
    for (int i = tid; i < Z_N - 1; i += 256) {
        float h  = sz[i + 1] - sz[i];
        float y0 = sB[i],  y1 = sB[i + 1];
        float d0 = sBz[i], d1 = sBz[i + 1];
        float c1 = d0;
        float c2 = (3.0f * (y1 - y0) / h - (2.0f * d0 + d1)) / h;
        float c3 = (2.0f * (y0 - y1) / h + (d0 + d1)) / (h * h);
        float av = c2 / c3;
        float bv = c1 / c3;
        float cv = (y0 - piv) / c3;

        float Qv  = (av * av - 3.0f * bv) / 9.0f;
        float Rr  = (2.0f * av * av * av - 9.0f * av * bv + 27.0f * cv) / 54.0f;
        float Q3  = Qv * Qv * Qv;
        bool three = (Rr * Rr) < Q3;
        float sqQ = sqrtf(Qv > 0.0f ? Qv : 1.0f);
        float den = three ? (sqQ * sqQ * sqQ) : 1.0f;
        float arg = fminf(fmaxf(Rr / den, -1.0f + 1e-7f), 1.0f - 1e-7f);
        float th  = acosf(three ? arg : 0.0f);
        float offc = -av / 3.0f;
        float disc = fmaxf(Rr * Rr - Q3, 0.0f);
        float sq   = sqrtf((three ? 1.0f : disc) + 1e-30f);
        float sgnR = (Rr > 0.0f) ? 1.0f : ((Rr < 0.0f) ? -1.0f : 0.0f);
        float Acb  = -sgnR * cbrtf(fabsf(Rr) + sq);
        bool okA   = fabsf(Acb) > 1e-12f;
        float Bcb  = okA ? (Qv / Acb) : 0.0f;
        float r1   = Acb + Bcb + offc;
        float t0 = -2.0f * sqQ * cosf(th / 3.0f) + offc;
        float t1 = -2.0f * sqQ * cosf((th + 2.0f * PI_F) / 3.0f) + offc;
        float t2 = -2.0f * sqQ * cosf((th + 4.0f * PI_F) / 3.0f) + offc;

        float roots[3];
        roots[0] = three ? t0 : r1;
        roots[1] = three ? t1 : BIGF;
        roots[2] = three ? t2 : BIGF;
        #pragma unroll
        for (int k = 0; k < 3; ++k) {
            float t = roots[k];
            if (t >= 0.0f && t < h) {
                float dB = (3.0f * c3 * t + 2.0f * c2) * t + c1;
                int pos = atomicAdd(&nRoots, 1);
                if (pos < CAP) { rz[pos] = sz[i] + t; rd[pos] = dB; }
            }
        }
    }
    __syncthreads();
    int nr = nRoots; if (nr > CAP) nr = CAP;
    for (int j = tid; j < CAP; j += 256) {
        if (j >= nr) { rz[j] = BIGF; rd[j] = 0.0f; }
    }

    // ---- Phase 4: bitonic sort of CAP=512 (key rz asc, payload rd) ---------
    for (unsigned k = 2; k <= CAP; k <<= 1) {
        for (unsigned j = k >> 1; j > 0; j >>= 1) {
            __syncthreads();
            for (unsigned t = tid; t < CAP; t += 256) {
                unsigned ixj = t ^ j;
                if (ixj > t) {
                    bool up = ((t & k) == 0);
                    float za = rz[t], zb = rz[ixj];
                    if (up ? (za > zb) : (za < zb)) {
                        float da = rd[t], db = rd[ixj];
                        rz[t] = zb; rz[ixj] = za;
                        rd[t] = db; rd[ixj] = da;
                    }
                }
            }
        }
    }
    __syncthreads();

    // ---- Phase 5: streaming classification (matches is1/is2/cumsum) --------
    if (tid == 0) {
        int c1n = 0, c2n = 0;
        for (int i = 0; i < CAP; ++i) {
            float zv = rz[i];
            if (zv >= BIGF * 0.5f) break;       // sorted: rest are padding
            float dB = rd[i];
            if (dB <= 0.0f) {
                if (c1n < W_N) z1s[c1n++] = zv;
            } else if (c1n > 0) {
                if (c2n < W_N) z2s[c2n++] = zv;
            }
        }
    }
    __syncthreads();
    if (tid < W_N) {
        float z1v = z1s[tid], z2v = z2s[tid];
        bool ok = (z2v > z1v) && (z2v < BIGF * 0.5f);
        if (!ok) { z1s[tid] = sz[0]; z2s[tid] = sz[0]; }
    }
    __syncthreads();

    // ---- Phase 6: quadrature integrand G[w][q] = f/bq ----------------------
    for (int task = tid; task < W_N * Q_N; task += 256) {
        int w = task >> 5;
        int q = task & 31;
        float z1v = z1s[w], z2v = z2s[w];
        float jac = 0.5f * (z2v - z1v);
        float zq  = z1v + jac * (sxs[q] + 1.0f);

        int i0 = (int)(zq * invh0);
        if (i0 < 0) i0 = 0;
        if (i0 > Z_N - 2) i0 = Z_N - 2;
        while (i0 > 0 && zq < sz[i0]) --i0;
        while (i0 < Z_N - 2 && zq >= sz[i0 + 1]) ++i0;
        float h = sz[i0 + 1] - sz[i0];
        float t = zq - sz[i0];

        // B spline
        float y0 = sB[i0], y1 = sB[i0 + 1], d0 = sBz[i0], d1 = sBz[i0 + 1];
        float c2 = (3.0f * (y1 - y0) / h - (2.0f * d0 + d1)) / h;
        float c3 = (2.0f * (y0 - y1) / h + (d0 + d1)) / (h * h);
        float Bq = ((c3 * t + c2) * t + d0) * t + y0;
        // b spline
        float e0 = sbv[i0], e1 = sbv[i0 + 1], f0 = sbz[i0], f1 = sbz[i0 + 1];
        float g2 = (3.0f * (e1 - e0) / h - (2.0f * f0 + f1)) / h;
        float g3 = (2.0f * (e0 - e1) / h + (f0 + f1)) / (h * h);
        float bq = ((g3 * t + g2) * t + f0) * t + e0;

        float rr = 1.0f - Bq / piv;
        float fv = (rr > 1e-12f) ? sqrtf(rr) : 0.0f;
        Gmat[w * Q_N + q] = fv / bq;
    }
    __syncthreads();

    // ---- Phase 7: out[w] = jac[w] * sum_q G[w,q]*ws[q] via WMMA f32 --------
    // V_WMMA_F32_16X16X4_F32: A = 16x4 tile of G (rows=w, cols=q-chunk),
    // B[k][n] = ws[k] broadcast across all N columns -> every column of D
    // holds the desired reduction. 8 K-chunks x 2 row-halves = 16 WMMAs.
    if (tid < 32) {
        const int lane = tid;
        const int khi  = (lane >> 4) * 2;        // lanes 16-31 carry K=2,3
        const int mrow = lane & 15;
        const int outBase = ((r * A_N + a) * P_N + p) * W_N;
        #pragma unroll
        for (int half = 0; half < 2; ++half) {
            const int wrow = half * 16 + mrow;
            v8f acc = {};
            #pragma unroll
            for (int k0 = 0; k0 < Q_N; k0 += 4) {
                v2f Aop, Bop;
                Aop.x = Gmat[wrow * Q_N + k0 + khi];
                Aop.y = Gmat[wrow * Q_N + k0 + khi + 1];
                Bop.x = sws[k0 + khi];
                Bop.y = sws[k0 + khi + 1];
                acc = __builtin_amdgcn_wmma_f32_16x16x4_f32(
                    /*neg_a=*/false, Aop, /*neg_b=*/false, Bop,
                    /*c_mod=*/(short)0, acc, /*reuse_a=*/false, /*reuse_b=*/false);
            }
            // D layout: lane 0 -> (M=0..7, N=0), lane 16 -> (M=8..15, N=0).
            if (mrow == 0) {
                const int wbase = half * 16 + ((lane >> 4) ? 8 : 0);
                #pragma unroll
                for (int jj = 0; jj < 8; ++jj) {
                    int w = wbase + jj;
                    float jacw = 0.5f * (z2s[w] - z1s[w]);
                    out[outBase + w] = acc[jj] * jacw;
                }
            }
        }
    }
}

extern "C" void kernel_launch(void* const* d_in, const int* in_sizes, int n_in,
                              void* d_out, int out_size, void* d_ws, size_t ws_size,
                              hipStream_t stream) {
    (void)in_sizes; (void)n_in; (void)out_size; (void)ws_size;
    const float* zeta   = (const float*)d_in[0];
    const float* B      = (const float*)d_in[1];
    const float* B_z    = (const float*)d_in[2];
    const float* Bsup   = (const float*)d_in[3];
    const float* Bsup_z = (const float*)d_in[4];
    float* glbuf = (float*)d_ws;                 // 64 floats: xs | ws
    float* outp  = (float*)d_out;

    gl_setup_kernel<<<1, 32, 0, stream>>>(glbuf);
    bounce_kernel<<<R_N * A_N * P_N, 256, 0, stream>>>(
        zeta, B, B_z, Bsup, Bsup_z, glbuf, outp);
}